// InterpretableConvLayer_83451214561432
// MI455X (gfx1250) — compile-verified
//
#include <hip/hip_runtime.h>
#include <hip/hip_bf16.h>

// ---------------------------------------------------------------------------
// InterpretableConvLayer on MI455X (gfx1250, wave32)
//   Pass 0: zero the 8192-entry u64 argmax-reduction buffer in d_ws
//   Pass 1: implicit-GEMM conv (V_WMMA_F32_16X16X4_F32) + bias + ReLU
//           -> writes y to d_out, fused packed-u64 argmax via LDS + global
//              atomic max per (batch, cout). K-loop address math replaced by
//              an LDS offset table (ds_load_b64 broadcast) + per-lane halo mask.
//   Pass 2: decode argmax index, gather template row, out = relu(y * t)
// ---------------------------------------------------------------------------

#define BATCH 128
#define CIN   32
#define COUT  64
#define NSP   56
#define HW    (NSP * NSP)          // 3136 spatial positions
#define KDIM  (CIN * 9)            // 288 = GEMM K
#define MCHUNKS (HW / 64)          // 49 M-chunks of 64 rows per batch

// ktab packing: bits[21:0] = byte offset into x-slab + 228 (bias for tap (0,0))
//               bits[25:22] = tap id r = kh*3+kw  (0..8)
#define KT_OFF_MASK 0x3FFFFF
#define KT_OFF_BIAS 228
#define KT_R_SHIFT  22

typedef float v2f __attribute__((ext_vector_type(2)));
typedef float v8f __attribute__((ext_vector_type(8)));

// ---------------------------------------------------------------------------
// Pass 0: zero reduction buffer (harness poisons ws once, never re-poisons)
// ---------------------------------------------------------------------------
__global__ void icl_init_red(unsigned long long* __restrict__ red) {
    int i = blockIdx.x * blockDim.x + threadIdx.x;
    if (i < BATCH * COUT) red[i] = 0ULL;
}

// A-element fetch: LDS table entry -> byte offset + tap id; halo-mask predicated
__device__ __forceinline__ float icl_fetch(const char* __restrict__ xpos,
                                           int e, unsigned vmask) {
    int boff = (e & KT_OFF_MASK) - KT_OFF_BIAS;
    int r    = e >> KT_R_SHIFT;
    float v  = 0.0f;
    if ((vmask >> r) & 1u)
        v = *(const float*)(xpos + boff);
    return v;
}

// ---------------------------------------------------------------------------
// Pass 1: conv+bias+relu via FP32 WMMA, fused packed argmax reduction
//   grid.x = BATCH * MCHUNKS, block = 128 threads (4 waves)
//   wave w owns M-rows [chunk*64 + w*16, +16), loops all 4 N-tiles (COUT=64)
// ---------------------------------------------------------------------------
__global__ void __launch_bounds__(128)
icl_conv_wmma(const float* __restrict__ x,        // [B, CIN, 56, 56]
              const float* __restrict__ Wt,       // [COUT, CIN, 3, 3] = [COUT, 288]
              const float* __restrict__ bias,     // [COUT]
              float* __restrict__ y,              // [B, COUT, 3136] (= d_out)
              unsigned long long* __restrict__ red) {  // [B, COUT]
    __shared__ int ktab[KDIM];
    __shared__ unsigned long long sred[COUT];

    // Build the k -> (byte offset, tap id) table once per block.
    for (int k = threadIdx.x; k < KDIM; k += 128) {
        int ci = k / 9;
        int r  = k - ci * 9;
        int kh = r / 3;
        int kw = r - kh * 3;
        int boff = 4 * (ci * HW + (kh - 1) * NSP + (kw - 1));
        ktab[k] = (boff + KT_OFF_BIAS) | (r << KT_R_SHIFT);
    }
    if (threadIdx.x < COUT) sred[threadIdx.x] = 0ULL;
    __syncthreads();

    const int chunk = blockIdx.x % MCHUNKS;
    const int b     = blockIdx.x / MCHUNKS;
    const int lane  = threadIdx.x & 31;
    const int wave  = threadIdx.x >> 5;

    const int m0   = chunk * 64 + wave * 16;
    const int mrow = m0 + (lane & 15);          // A-matrix row for this lane
    const int oh   = mrow / NSP;
    const int ow   = mrow - oh * NSP;
    const int khalf = (lane >> 4) << 1;         // lanes 0-15: K+0..1, 16-31: K+2..3
    const int nlo   = lane & 15;                // B/C/D column within tile

    // Per-lane pointer at this output position's (0,0,center) in the x slab
    const char* xpos = (const char*)(x + (size_t)b * CIN * HW + oh * NSP + ow);

    // Per-lane 9-bit halo validity mask (taps r = kh*3+kw)
    unsigned vmask = 0;
    #pragma unroll
    for (int r = 0; r < 9; ++r) {
        const int kh = r / 3, kw = r % 3;       // compile-time constants
        const int ih = oh + kh - 1;
        const int iw = ow + kw - 1;
        if (((unsigned)ih < (unsigned)NSP) && ((unsigned)iw < (unsigned)NSP))
            vmask |= (1u << r);
    }

    v8f acc[4] = {};                            // 4 N-tiles of 16x16 f32

    #pragma unroll 4
    for (int kb = 0; kb < KDIM / 4; ++kb) {
        const int kbase = kb * 4 + khalf;
        const int2 e = *(const int2*)(&ktab[kbase]);   // ds_load_b64 broadcast
        v2f av;
        av[0] = icl_fetch(xpos, e.x, vmask);
        av[1] = icl_fetch(xpos, e.y, vmask);
        #pragma unroll
        for (int nt = 0; nt < 4; ++nt) {
            const float* wp = Wt + (size_t)(nt * 16 + nlo) * KDIM + kbase;
            v2f bv;
            bv[0] = wp[0];
            bv[1] = wp[1];
            // D = A(16x4,f32) * B(4x16,f32) + C  -> v_wmma_f32_16x16x4_f32
            acc[nt] = __builtin_amdgcn_wmma_f32_16x16x4_f32(
                false, av, false, bv, (short)0, acc[nt], false, false);
        }
    }

    // ---- epilogue: bias + relu + store y + packed argmax reduction ----
    const int mhi = (lane >> 4) << 3;           // C/D layout: +8 rows for lanes>=16
    #pragma unroll
    for (int nt = 0; nt < 4; ++nt) {
        const int n = nt * 16 + nlo;
        const float bn = bias[n];
        float* yrow = y + ((size_t)b * COUT + n) * HW;
        unsigned long long best = 0ULL;
        #pragma unroll
        for (int v = 0; v < 8; ++v) {
            const int M = m0 + mhi + v;
            float yv = acc[nt][v] + bn;
            yv = yv > 0.0f ? yv : 0.0f;
            yrow[M] = yv;
            // pack: value in high 32 (monotone bits since yv >= 0),
            // ~index in low 32 so ties resolve to the LOWEST index (jnp.argmax)
            unsigned long long pk =
                ((unsigned long long)__float_as_uint(yv) << 32) |
                (unsigned long long)(0xFFFFFFFFu - (unsigned)M);
            best = pk > best ? pk : best;
        }
        atomicMax(&sred[n], best);              // LDS u64 atomic max
    }
    __syncthreads();
    if (threadIdx.x < COUT)
        atomicMax(&red[b * COUT + threadIdx.x], sred[threadIdx.x]);
}

// ---------------------------------------------------------------------------
// Pass 2: out = relu(y * templates[argmax]), float4-vectorized, in place
// ---------------------------------------------------------------------------
__global__ void __launch_bounds__(256)
icl_mask_apply(float* __restrict__ y,                      // d_out, in place
               const float* __restrict__ templates,        // [3137, 3136]
               const unsigned long long* __restrict__ red) {
    const int t = blockIdx.x * blockDim.x + threadIdx.x;   // quad index
    const int total4 = BATCH * COUT * (HW / 4);
    if (t >= total4) return;
    const int base = t * 4;                                // never crosses a row
    const int m  = base % HW;
    const int bc = base / HW;                              // b*COUT + n

    const unsigned idx =
        0xFFFFFFFFu - (unsigned)(red[bc] & 0xFFFFFFFFull); // argmax spatial index

    const float4 tv = *(const float4*)(templates + (size_t)idx * HW + m);
    float4 yv = *(float4*)(y + base);
    float4 o;
    o.x = yv.x * tv.x; o.x = o.x > 0.0f ? o.x : 0.0f;
    o.y = yv.y * tv.y; o.y = o.y > 0.0f ? o.y : 0.0f;
    o.z = yv.z * tv.z; o.z = o.z > 0.0f ? o.z : 0.0f;
    o.w = yv.w * tv.w; o.w = o.w > 0.0f ? o.w : 0.0f;
    *(float4*)(y + base) = o;
}

// ---------------------------------------------------------------------------
extern "C" void kernel_launch(void* const* d_in, const int* in_sizes, int n_in,
                              void* d_out, int out_size, void* d_ws, size_t ws_size,
                              hipStream_t stream) {
    const float* x    = (const float*)d_in[0];   // [128, 32, 56, 56]
    const float* Wt   = (const float*)d_in[1];   // [64, 32, 3, 3]
    const float* bias = (const float*)d_in[2];   // [64]
    const float* tmpl = (const float*)d_in[3];   // [3137, 56, 56]
    float* out = (float*)d_out;                  // [128, 64, 56, 56]
    unsigned long long* red = (unsigned long long*)d_ws;   // 8192 * 8 B = 64 KB

    // Pass 0: zero the per-(b,c) packed-argmax buffer
    icl_init_red<<<(BATCH * COUT + 255) / 256, 256, 0, stream>>>(red);

    // Pass 1: WMMA conv + bias + relu + fused argmax
    icl_conv_wmma<<<BATCH * MCHUNKS, 128, 0, stream>>>(x, Wt, bias, out, red);

    // Pass 2: template gather + mask + relu (in place on d_out)
    const int total4 = BATCH * COUT * (HW / 4);
    icl_mask_apply<<<(total4 + 255) / 256, 256, 0, stream>>>(out, tmpl, red);
}